// L1Embedding_11802570129618
// MI455X (gfx1250) — compile-verified
//
#include <hip/hip_runtime.h>
#include <hip/hip_bf16.h>

#define DIMS 64
#define NVEC 1024
#define NROWS 4096
#define ROWS_PER_WG 32

typedef __attribute__((ext_vector_type(2))) float v2f;
typedef __attribute__((ext_vector_type(8))) float v8f;

// ---------------------------------------------------------------------------
// Kernel 0: re-initialize the min workspaces to +inf bits every call
// ---------------------------------------------------------------------------
__global__ void init_mins_kernel(unsigned* __restrict__ rowmin,
                                 unsigned* __restrict__ colmin) {
    int i = blockIdx.x * blockDim.x + threadIdx.x;
    if (i < NROWS) rowmin[i] = 0x7F800000u;  // +inf
    if (i < NVEC)  colmin[i] = 0x7F800000u;
}

// ---------------------------------------------------------------------------
// Kernel 1: L1 distance matrix + row/col running mins.
//   lane  <-> one codebook column v (p[v][0..63] held in 64 VGPRs)
//   wave  <-> 32 columns x ROWS_PER_WG rows
//   x[row][*] is wave-uniform -> scalarized to s_load (constant cache),
//   inner loop is pure VALU: v_sub_f32 + v_add_f32 |abs| per (b,v,d).
// ---------------------------------------------------------------------------
__global__ __launch_bounds__(256) void
l1_dist_kernel(const float* __restrict__ x, const float* __restrict__ p,
               float* __restrict__ dist,
               unsigned* __restrict__ rowmin, unsigned* __restrict__ colmin) {
    const int lane = threadIdx.x & 31;
    const int wave = threadIdx.x >> 5;
    const int v = blockIdx.y * 256 + wave * 32 + lane;  // gridDim.y = 4
    const int rowBase = blockIdx.x * ROWS_PER_WG;       // gridDim.x = 128

    // Load this lane's codebook column into registers (one time).
    float pv[DIMS];
#pragma unroll
    for (int d = 0; d < DIMS; d += 4) {
        float4 t = *(const float4*)(p + (size_t)v * DIMS + d);
        pv[d + 0] = t.x; pv[d + 1] = t.y; pv[d + 2] = t.z; pv[d + 3] = t.w;
    }

    float colAcc = __builtin_inff();  // running min over rows for column v

    for (int r = 0; r < ROWS_PER_WG; ++r) {
        const int row = rowBase + r;
        const float* xrow = x + (size_t)row * DIMS;  // uniform address -> s_load
        // gfx1250 prefetch of the next row's cacheline(s)
        __builtin_prefetch(xrow + DIMS, 0, 0);

        float acc = 0.0f;
#pragma unroll
        for (int d = 0; d < DIMS; ++d)
            acc += __builtin_fabsf(xrow[d] - pv[d]);  // sub + add|.| : 2 VALU

        dist[(size_t)row * NVEC + v] = acc;
        colAcc = fminf(colAcc, acc);

        // wave32 min-reduce across the 32 columns of this wave -> row partial
        float m = acc;
#pragma unroll
        for (int off = 16; off > 0; off >>= 1)
            m = fminf(m, __shfl_xor(m, off, 32));
        if (lane == 0)
            atomicMin(rowmin + row, __float_as_uint(m));  // dist>=0: uint order ok
    }
    atomicMin(colmin + v, __float_as_uint(colAcc));
}

// ---------------------------------------------------------------------------
// Kernel 2: means of the min arrays via V_WMMA_F32_16X16X4_F32 with B = ones.
//   D[i][j] = sum_k A[i][k]  =>  sum over all of D = 16 * sum(A).
//   One wave (32 lanes), EXEC all ones as WMMA requires.
// ---------------------------------------------------------------------------
__global__ __launch_bounds__(32) void
means_kernel(const unsigned* __restrict__ rowmin,
             const unsigned* __restrict__ colmin, float* __restrict__ out) {
    const int lane = threadIdx.x;
    v2f ones; ones.x = 1.0f; ones.y = 1.0f;

    // r2 = mean over 4096 row-mins
    v8f acc = {};
    for (int i = 0; i < NROWS; i += 64) {
        v2f a;
        a.x = __uint_as_float(rowmin[i + 2 * lane + 0]);
        a.y = __uint_as_float(rowmin[i + 2 * lane + 1]);
        acc = __builtin_amdgcn_wmma_f32_16x16x4_f32(
            false, a, false, ones, (short)0, acc, false, false);
    }
    float s = 0.0f;
#pragma unroll
    for (int k = 0; k < 8; ++k) s += acc[k];
#pragma unroll
    for (int off = 16; off > 0; off >>= 1) s += __shfl_xor(s, off, 32);
    const float r2 = s * (1.0f / 16.0f) * (1.0f / (float)NROWS);

    // r1 = mean over 1024 col-mins
    v8f acc2 = {};
    for (int i = 0; i < NVEC; i += 64) {
        v2f a;
        a.x = __uint_as_float(colmin[i + 2 * lane + 0]);
        a.y = __uint_as_float(colmin[i + 2 * lane + 1]);
        acc2 = __builtin_amdgcn_wmma_f32_16x16x4_f32(
            false, a, false, ones, (short)0, acc2, false, false);
    }
    float s2 = 0.0f;
#pragma unroll
    for (int k = 0; k < 8; ++k) s2 += acc2[k];
#pragma unroll
    for (int off = 16; off > 0; off >>= 1) s2 += __shfl_xor(s2, off, 32);
    const float r1 = s2 * (1.0f / 16.0f) * (1.0f / (float)NVEC);

    if (lane == 0) { out[0] = r1; out[1] = r2; }
}

// ---------------------------------------------------------------------------
extern "C" void kernel_launch(void* const* d_in, const int* in_sizes, int n_in,
                              void* d_out, int out_size, void* d_ws, size_t ws_size,
                              hipStream_t stream) {
    (void)in_sizes; (void)n_in; (void)out_size; (void)ws_size;
    const float* x = (const float*)d_in[0];   // (4096, 64)
    const float* p = (const float*)d_in[1];   // (1024, 64)
    float* dist = (float*)d_out;              // (4096*1024) then r1, r2

    unsigned* rowmin = (unsigned*)d_ws;       // 4096
    unsigned* colmin = rowmin + NROWS;        // 1024

    init_mins_kernel<<<(NROWS + 255) / 256, 256, 0, stream>>>(rowmin, colmin);

    dim3 grid(NROWS / ROWS_PER_WG, NVEC / 256);  // 128 x 4
    l1_dist_kernel<<<grid, 256, 0, stream>>>(x, p, dist, rowmin, colmin);

    means_kernel<<<1, 32, 0, stream>>>(rowmin, colmin,
                                       dist + (size_t)NROWS * NVEC);
}